// TAdaConv2dV2_10917806866926
// MI455X (gfx1250) — compile-verified
//
#include <hip/hip_runtime.h>

typedef float v2f __attribute__((ext_vector_type(2)));
typedef float v8f __attribute__((ext_vector_type(8)));

#define CC   256
#define CRR  64
#define TT   8
#define BB   8
#define HH   56
#define WWD  56
#define HWP  (HH*WWD)      // 3136
#define BT   (BB*TT)       // 64

#ifndef USE_ASYNC_LDS
#define USE_ASYNC_LDS 1
#endif

// ---------------- K1: spatial average pool  feat[b,c,t,:,:] -> x0[b,c,t] ----
__global__ __launch_bounds__(256) void k_pool(const float* __restrict__ feat,
                                              float* __restrict__ x0) {
    int bct = blockIdx.x;                       // (b*C + c)*T + t
    const float* src = feat + (size_t)bct * HWP;
    float s = 0.f;
    for (int i = threadIdx.x; i < HWP; i += 256) s += src[i];
    __shared__ float red[256];
    red[threadIdx.x] = s; __syncthreads();
    for (int off = 128; off > 0; off >>= 1) {
        if (threadIdx.x < off) red[threadIdx.x] += red[threadIdx.x + off];
        __syncthreads();
    }
    if (threadIdx.x == 0) x0[bct] = red[0] * (1.0f / (float)HWP);
}

// ---------------- K2: weight reorder  W[co,ci,kh,kw] -> Wt[rs][ci][co] ------
__global__ __launch_bounds__(256) void k_wt(const float* __restrict__ Wsrc,
                                            float* __restrict__ Wt) {
    int i = blockIdx.x * 256 + threadIdx.x;     // 9*256*256
    if (i >= 9 * CC * CC) return;
    int co = i % CC;
    int ci = (i / CC) % CC;
    int rs = i / (CC * CC);
    Wt[i] = Wsrc[((size_t)co * CC + ci) * 9 + rs];
}

// ---------------- K3: routing function (conv-a, LN, qGELU, attn, conv-b) ----
__global__ __launch_bounds__(256) void k_route(
    const float* __restrict__ x0,
    const float* __restrict__ a_w,  const float* __restrict__ a_b,
    const float* __restrict__ nw,   const float* __restrict__ nb,
    const float* __restrict__ ntw,  const float* __restrict__ ntb,
    const float* __restrict__ qkvw, const float* __restrict__ qkvb,
    const float* __restrict__ aow,  const float* __restrict__ aob,
    const float* __restrict__ bw,   const float* __restrict__ bbw,
    const float* __restrict__ bias_p,
    float* __restrict__ walpha, float* __restrict__ biasbt) {
    int b = blockIdx.x, tid = threadIdx.x;
    __shared__ float xin[CC * TT];       // [c][t]
    __shared__ float xg [CRR * TT];      // [cr][t]  (conv-a out, then qgelu(ln))
    __shared__ float xn [CRR * TT];
    __shared__ float qkv[3 * CRR * TT];  // [row][t]
    __shared__ float att[TT * TT];
    __shared__ float ov [TT * CRR];      // [t][d]
    __shared__ float xf [CRR * TT];
    __shared__ float mu[TT], rsg[TT];

    for (int i = tid; i < CC * TT; i += 256) xin[i] = x0[(size_t)b * CC * TT + i];
    __syncthreads();

    // conv 'a': [CR,C,3] over t, pad 1, + bias
    for (int o = tid; o < CRR * TT; o += 256) {
        int cr = o >> 3, t = o & 7;
        float s = a_b[cr];
        const float* wr = a_w + (size_t)cr * CC * 3;
        for (int c = 0; c < CC; ++c) {
            const float* xr = xin + c * TT;
            float xm = (t > 0)      ? xr[t - 1] : 0.f;
            float xp = (t < TT - 1) ? xr[t + 1] : 0.f;
            s += wr[c * 3 + 0] * xm + wr[c * 3 + 1] * xr[t] + wr[c * 3 + 2] * xp;
        }
        xg[o] = s;
    }
    __syncthreads();
    // LN over channels, per t, then qGELU
    if (tid < TT) {
        float m = 0.f, v = 0.f;
        for (int cr = 0; cr < CRR; ++cr) m += xg[cr * TT + tid];
        m *= (1.f / CRR);
        for (int cr = 0; cr < CRR; ++cr) { float d = xg[cr * TT + tid] - m; v += d * d; }
        v *= (1.f / CRR);
        mu[tid] = m; rsg[tid] = rsqrtf(v + 1e-6f);
    }
    __syncthreads();
    for (int o = tid; o < CRR * TT; o += 256) {
        int cr = o >> 3, t = o & 7;
        float xv = (xg[o] - mu[t]) * rsg[t] * nw[cr] + nb[cr];
        xg[o] = xv / (1.f + __expf(-1.702f * xv));     // qGELU
    }
    __syncthreads();
    // LN_t
    if (tid < TT) {
        float m = 0.f, v = 0.f;
        for (int cr = 0; cr < CRR; ++cr) m += xg[cr * TT + tid];
        m *= (1.f / CRR);
        for (int cr = 0; cr < CRR; ++cr) { float d = xg[cr * TT + tid] - m; v += d * d; }
        v *= (1.f / CRR);
        mu[tid] = m; rsg[tid] = rsqrtf(v + 1e-6f);
    }
    __syncthreads();
    for (int o = tid; o < CRR * TT; o += 256) {
        int cr = o >> 3, t = o & 7;
        xn[o] = (xg[o] - mu[t]) * rsg[t] * ntw[cr] + ntb[cr];
    }
    __syncthreads();
    // qkv projection
    for (int o = tid; o < 3 * CRR * TT; o += 256) {
        int row = o >> 3, t = o & 7;
        float s = qkvb[row];
        const float* wr = qkvw + (size_t)row * CRR;
        for (int cr = 0; cr < CRR; ++cr) s += wr[cr] * xn[cr * TT + t];
        qkv[o] = s;
    }
    __syncthreads();
    // attention scores (nh = 1, scale = 64^-0.5)
    if (tid < TT * TT) {
        int tq = tid >> 3, tk = tid & 7;
        float s = 0.f;
        for (int d = 0; d < CRR; ++d) s += qkv[d * TT + tq] * qkv[(CRR + d) * TT + tk];
        att[tid] = s * 0.125f;
    }
    __syncthreads();
    if (tid < TT) {
        float mx = -1e30f;
        for (int k = 0; k < TT; ++k) mx = fmaxf(mx, att[tid * TT + k]);
        float ss = 0.f;
        for (int k = 0; k < TT; ++k) { float e = __expf(att[tid * TT + k] - mx); att[tid * TT + k] = e; ss += e; }
        float inv = 1.f / ss;
        for (int k = 0; k < TT; ++k) att[tid * TT + k] *= inv;
    }
    __syncthreads();
    for (int o = tid; o < TT * CRR; o += 256) {       // ov[t][d] = attn @ v
        int t = o / CRR, d = o % CRR;
        float s = 0.f;
        for (int tk = 0; tk < TT; ++tk) s += att[t * TT + tk] * qkv[(2 * CRR + d) * TT + tk];
        ov[o] = s;
    }
    __syncthreads();
    // output proj + residual
    for (int o = tid; o < CRR * TT; o += 256) {
        int cr = o >> 3, t = o & 7;
        float s = aob[cr];
        const float* wr = aow + (size_t)cr * CRR;
        for (int d = 0; d < CRR; ++d) s += wr[d] * ov[t * CRR + d];
        xf[o] = xg[o] + s;
    }
    __syncthreads();
    // w_alpha / b_alpha (conv1d kernel 3, pad 1, no bias) stored [bt][c]
    for (int o = tid; o < CC * TT; o += 256) {
        int c = o >> 3, t = o & 7;
        float sa = 0.f, sb = 0.f;
        const float* wa = bw  + (size_t)c * CRR * 3;
        const float* wb = bbw + (size_t)c * CRR * 3;
        for (int cr = 0; cr < CRR; ++cr) {
            float xm = (t > 0)      ? xf[cr * TT + t - 1] : 0.f;
            float xc = xf[cr * TT + t];
            float xp = (t < TT - 1) ? xf[cr * TT + t + 1] : 0.f;
            sa += wa[cr * 3] * xm + wa[cr * 3 + 1] * xc + wa[cr * 3 + 2] * xp;
            sb += wb[cr * 3] * xm + wb[cr * 3 + 1] * xc + wb[cr * 3 + 2] * xp;
        }
        size_t idx = ((size_t)b * TT + t) * CC + c;
        walpha[idx] = sa;
        biasbt[idx] = sb * bias_p[c];
    }
}

// ---------------- K4: calibrated 3x3 conv as implicit GEMM, f32 WMMA --------
// Block: 256 thr (8 wave32), 64(co) x 64(hw) tile, K chunked 32, 9 shifted GEMMs.
__global__ __launch_bounds__(256) void k_conv(
    const float* __restrict__ feat, const float* __restrict__ Wt,
    const float* __restrict__ walpha, const float* __restrict__ biasbt,
    float* __restrict__ cout_) {
    int bi   = blockIdx.x;
    int img  = bi / (4 * 49);                  // b*8 + t
    int rem  = bi % (4 * 49);
    int mblk = rem / 49;                       // co block of 64
    int nblk = rem % 49;                       // hw block of 64
    int b = img >> 3, t = img & 7;
    int co0 = mblk * 64;
    int p0  = nblk * 64;

    __shared__ float As[32 * 64];              // [k][m]  weights
    __shared__ float Bs[32 * 64];              // [k][n]  scaled shifted input
    __shared__ float wal[CC];

    int tid  = threadIdx.x;
    int lane = tid & 31, wave = tid >> 5;
    int co_off = (wave & 3) * 16;              // 4 waves across co
    int n_off  = (wave >> 2) * 32;             // 2 waves across hw
    int khalf  = lane >> 4;                    // 0: K0/K1, 1: K2/K3
    int lm     = lane & 15;

    for (int i = tid; i < CC; i += 256) wal[i] = walpha[(size_t)img * CC + i];

    v8f acc0 = {}; v8f acc1 = {};

    const float* fimg = feat + ((size_t)(b * CC) * TT + t) * HWP;  // +ci*T*HWP

    // this thread stages k-row lk, columns ln..ln+7 of both panels
    int lk = tid >> 3;                         // 0..31
    int ln = (tid & 7) * 8;                    // 0,8,...,56

    // precomputed geometry of the 8 B elements (fixed for whole kernel)
    int ph[8], pw[8], pp[8];
    #pragma unroll
    for (int j = 0; j < 8; ++j) {
        int p = p0 + ln + j;
        pp[j] = p; ph[j] = p / WWD; pw[j] = p % WWD;
    }

    for (int rs = 0; rs < 9; ++rs) {
        int r = rs / 3 - 1, s = rs % 3 - 1;
        int d = r * WWD + s;                    // linear shift when in-row
        const float* wtp = Wt + (size_t)rs * CC * CC;
        for (int ci0 = 0; ci0 < CC; ci0 += 32) {
            // ---- A panel: As[k][m] = Wt[rs][(ci0+k)*C + co0+m], 32B/thread
            const float* ga = wtp + (size_t)(ci0 + lk) * CC + co0 + ln;
#if USE_ASYNC_LDS
            {
                unsigned dst = (unsigned)(size_t)(void*)(As + lk * 64 + ln);
                asm volatile(
                    "global_load_async_to_lds_b128 %0, %1, off\n\t"
                    "global_load_async_to_lds_b128 %2, %3, off"
                    :: "v"(dst), "v"(ga), "v"(dst + 16u), "v"(ga + 4)
                    : "memory");
            }
#else
            {
                float4 a0 = *(const float4*)(ga);
                float4 a1 = *(const float4*)(ga + 4);
                *(float4*)(As + lk * 64 + ln)     = a0;
                *(float4*)(As + lk * 64 + ln + 4) = a1;
            }
#endif
            // ---- B panel: branchless masked load (pad folded into scale)
            {
                int ci = ci0 + lk;
                const float* fsl = fimg + (size_t)ci * ((size_t)TT * HWP);
                float scale = wal[ci];
                #pragma unroll
                for (int j = 0; j < 8; ++j) {
                    bool valid = ((unsigned)(ph[j] + r) < (unsigned)HH) &&
                                 ((unsigned)(pw[j] + s) < (unsigned)WWD);
                    int raw = pp[j] + d;
                    int idx = raw < 0 ? 0 : (raw > HWP - 1 ? HWP - 1 : raw);
                    Bs[lk * 64 + ln + j] = fsl[idx] * (valid ? scale : 0.f);
                }
            }
#if USE_ASYNC_LDS
            asm volatile("s_wait_asynccnt 0" ::: "memory");
#endif
            __syncthreads();
            #pragma unroll
            for (int kk = 0; kk < 8; ++kk) {
                int kb = kk * 4 + khalf * 2;
                v2f a, b0, b1;
                a.x  = As[kb * 64 + co_off + lm];
                a.y  = As[(kb + 1) * 64 + co_off + lm];
                b0.x = Bs[kb * 64 + n_off + lm];
                b0.y = Bs[(kb + 1) * 64 + n_off + lm];
                b1.x = Bs[kb * 64 + n_off + 16 + lm];
                b1.y = Bs[(kb + 1) * 64 + n_off + 16 + lm];
                acc0 = __builtin_amdgcn_wmma_f32_16x16x4_f32(
                           false, a, false, b0, (short)0, acc0, false, false);
                acc1 = __builtin_amdgcn_wmma_f32_16x16x4_f32(
                           false, a, false, b1, (short)0, acc1, false, false);
            }
            __syncthreads();
        }
    }
    // epilogue: + per-(bt,co) bias, store [img][co][p]
    float* obase = cout_ + ((size_t)img * CC + co0) * HWP + p0;
    #pragma unroll
    for (int i = 0; i < 8; ++i) {
        int m = co_off + khalf * 8 + i;
        float bias = biasbt[(size_t)img * CC + co0 + m];
        obase[(size_t)m * HWP + n_off + lm]      = acc0[i] + bias;
        obase[(size_t)m * HWP + n_off + 16 + lm] = acc1[i] + bias;
    }
}

// ---------------- K5: per-channel BN statistics (out and t-avgpool) ---------
__global__ __launch_bounds__(256) void k_zero(float* __restrict__ p, int n) {
    int i = blockIdx.x * 256 + threadIdx.x;
    if (i < n) p[i] = 0.f;
}

__global__ __launch_bounds__(256) void k_stats(const float* __restrict__ co_,
                                               float* __restrict__ stats) {
    int blk = blockIdx.x;                    // c*BT + img
    int c = blk / BT, img = blk % BT;
    int t = img & 7;
    const float* base = co_ + ((size_t)img * CC + c) * HWP;
    const float* bm = (t > 0)      ? base - (size_t)CC * HWP : nullptr;
    const float* bp = (t < TT - 1) ? base + (size_t)CC * HWP : nullptr;
    float s1 = 0, s2 = 0, s3 = 0, s4 = 0;
    for (int p = threadIdx.x; p < HWP; p += 256) {
        float v = base[p];
        float a = (v + (bm ? bm[p] : 0.f) + (bp ? bp[p] : 0.f)) * (1.f / 3.f);
        s1 += v; s2 += v * v; s3 += a; s4 += a * a;
    }
    __shared__ float r1[256], r2[256], r3[256], r4[256];
    int tid = threadIdx.x;
    r1[tid] = s1; r2[tid] = s2; r3[tid] = s3; r4[tid] = s4;
    __syncthreads();
    for (int off = 128; off > 0; off >>= 1) {
        if (tid < off) {
            r1[tid] += r1[tid + off]; r2[tid] += r2[tid + off];
            r3[tid] += r3[tid + off]; r4[tid] += r4[tid + off];
        }
        __syncthreads();
    }
    if (tid == 0) {
        atomicAdd(&stats[c],          r1[0]);
        atomicAdd(&stats[CC + c],     r2[0]);
        atomicAdd(&stats[2 * CC + c], r3[0]);
        atomicAdd(&stats[3 * CC + c], r4[0]);
    }
}

// ---------------- K6: dual BN + layout transpose to [b,c,t,h,w] -------------
__global__ __launch_bounds__(256) void k_final(
    const float* __restrict__ co_, const float* __restrict__ stats,
    const float* __restrict__ ga, const float* __restrict__ ba,
    const float* __restrict__ gb, const float* __restrict__ bb_,
    float* __restrict__ out) {
    size_t gid = (size_t)blockIdx.x * 256 + threadIdx.x;
    if (gid >= (size_t)BT * CC * HWP) return;
    int p = (int)(gid % HWP);
    size_t q = gid / HWP;
    int c = (int)(q % CC);
    int img = (int)(q / CC);
    int b = img >> 3, t = img & 7;
    float v  = co_[gid];
    float vm = (t > 0)      ? co_[gid - (size_t)CC * HWP] : 0.f;
    float vp = (t < TT - 1) ? co_[gid + (size_t)CC * HWP] : 0.f;
    float a = (v + vm + vp) * (1.f / 3.f);
    const float Ninv = 1.f / (float)((size_t)BB * TT * HWP);
    float mu_a  = stats[c] * Ninv;
    float var_a = stats[CC + c] * Ninv - mu_a * mu_a;
    float mu_b  = stats[2 * CC + c] * Ninv;
    float var_b = stats[3 * CC + c] * Ninv - mu_b * mu_b;
    float r = ga[c] * (v - mu_a) * rsqrtf(var_a + 1e-5f) + ba[c]
            + gb[c] * (a - mu_b) * rsqrtf(var_b + 1e-5f) + bb_[c];
    out[(((size_t)b * CC + c) * TT + t) * HWP + p] = r;
}

// ---------------------------------------------------------------------------
extern "C" void kernel_launch(void* const* d_in, const int* in_sizes, int n_in,
                              void* d_out, int out_size, void* d_ws, size_t ws_size,
                              hipStream_t stream) {
    const float* feat   = (const float*)d_in[0];
    const float* W      = (const float*)d_in[1];
    const float* bias_p = (const float*)d_in[2];
    const float* a_w    = (const float*)d_in[3];
    const float* a_b    = (const float*)d_in[4];
    const float* norm_w = (const float*)d_in[5];
    const float* norm_b = (const float*)d_in[6];
    const float* normtw = (const float*)d_in[7];
    const float* normtb = (const float*)d_in[8];
    const float* qkv_w  = (const float*)d_in[9];
    const float* qkv_b  = (const float*)d_in[10];
    const float* ao_w   = (const float*)d_in[11];
    const float* ao_b   = (const float*)d_in[12];
    const float* b_w    = (const float*)d_in[13];
    const float* bb_w   = (const float*)d_in[14];
    const float* gamma_a= (const float*)d_in[15];
    const float* beta_a = (const float*)d_in[16];
    const float* gamma_b= (const float*)d_in[17];
    const float* beta_b = (const float*)d_in[18];
    float* out = (float*)d_out;

    // workspace carve-up (floats)
    float* ws = (float*)d_ws;
    float* x0      = ws;                       // 16384
    float* walpha  = x0 + 16384;               // 16384
    float* biasbt  = walpha + 16384;           // 16384
    float* Wt      = biasbt + 16384;           // 589824
    float* stats   = Wt + 589824;              // 1024
    float* convout = stats + 1024;             // 51,380,224

    k_pool <<<BB * CC * TT, 256, 0, stream>>>(feat, x0);
    k_wt   <<<(9 * CC * CC + 255) / 256, 256, 0, stream>>>(W, Wt);
    k_route<<<BB, 256, 0, stream>>>(x0, a_w, a_b, norm_w, norm_b, normtw, normtb,
                                    qkv_w, qkv_b, ao_w, ao_b, b_w, bb_w, bias_p,
                                    walpha, biasbt);
    k_conv <<<BT * 4 * 49, 256, 0, stream>>>(feat, Wt, walpha, biasbt, convout);
    k_zero <<<4, 256, 0, stream>>>(stats, 4 * CC);
    k_stats<<<CC * BT, 256, 0, stream>>>(convout, stats);
    size_t total = (size_t)BT * CC * HWP;
    k_final<<<(unsigned)((total + 255) / 256), 256, 0, stream>>>(
        convout, stats, gamma_a, beta_a, gamma_b, beta_b, out);
}